// KSparse_52879637348503
// MI455X (gfx1250) — compile-verified
//
#include <hip/hip_runtime.h>
#include <stdint.h>

#define N_FEAT 8192
#define BLOCK  256
#define NVEC   (N_FEAT / 4)     // 2048 float4 per row
#define ITERS  (NVEC / BLOCK)   // 8 float4 per thread

typedef int v4i __attribute__((vector_size(16)));
typedef __attribute__((address_space(1))) v4i glob_v4i;
typedef __attribute__((address_space(3))) v4i lds_v4i;

#if defined(__gfx1250__) && __has_builtin(__builtin_amdgcn_global_load_async_to_lds_b128)
#define HAVE_ASYNC_BUILTIN 1
#else
#define HAVE_ASYNC_BUILTIN 0
#endif

static __device__ __forceinline__ void async_load_b128_to_lds(const float* g, void* l) {
    // low 32 bits of a generic LDS address are the LDS byte offset (AS3 value)
    uint32_t loff = (uint32_t)(uintptr_t)l;
#if HAVE_ASYNC_BUILTIN
    __builtin_amdgcn_global_load_async_to_lds_b128(
        (glob_v4i*)(uintptr_t)g, (lds_v4i*)(uintptr_t)loff, 0, 0);
#else
    asm volatile("global_load_async_to_lds_b128 %0, %1, off"
                 :: "v"(loff), "v"(g) : "memory");
#endif
}

static __device__ __forceinline__ void wait_async_zero() {
#if __has_builtin(__builtin_amdgcn_s_wait_asynccnt)
    __builtin_amdgcn_s_wait_asynccnt(0);
#else
    asm volatile("s_wait_asynccnt 0" ::: "memory");
#endif
}

// Order-preserving float -> u32 key (ascending float order == ascending key order)
static __device__ __forceinline__ unsigned key_of(float f) {
    unsigned u = __float_as_uint(f);
    return u ^ ((unsigned)((int)u >> 31) | 0x80000000u);
}

__global__ __launch_bounds__(BLOCK)
void KSparse_52879637348503_kernel(const float* __restrict__ in,
                                   const int* __restrict__ kptr,
                                   float* __restrict__ out) {
    __shared__ __align__(16) float row[N_FEAT];        // 32 KB
    __shared__ unsigned hist[256];                     // 1 KB
    __shared__ unsigned sh_prefix;
    __shared__ unsigned sh_rank;

    const int tid = threadIdx.x;
    const size_t rbase = (size_t)blockIdx.x * N_FEAT;
    const float* __restrict__ rowp = in + rbase;

    // ---- Stage row into LDS via CDNA5 async global->LDS datapath ----
    #pragma unroll
    for (int i = 0; i < ITERS; ++i) {
        const int v = tid + i * BLOCK;                 // float4 index
        async_load_b128_to_lds(rowp + v * 4, &row[v * 4]);
    }
    wait_async_zero();

    const int kk = *kptr;                              // 512 in this problem
    const int actual_k = (kk < N_FEAT - 1) ? kk : (N_FEAT - 1);
    if (tid == 0) {
        sh_prefix = 0u;
        sh_rank = (unsigned)actual_k;                  // descending-rank target
    }
    __syncthreads();

    // ---- 4-pass MSD radix select: recover exact key of rank-k element ----
    #pragma unroll
    for (int pass = 0; pass < 4; ++pass) {
        hist[tid & 255] = 0u;
        __syncthreads();

        const int shift = 24 - 8 * pass;
        const unsigned pmask = (pass == 0) ? 0u : (0xFFFFFFFFu << (shift + 8));
        const unsigned pref = sh_prefix;

        for (int i = tid; i < N_FEAT; i += BLOCK) {
            unsigned key = key_of(row[i]);
            if ((key & pmask) == pref)
                atomicAdd(&hist[(key >> shift) & 255u], 1u);
        }
        __syncthreads();

        if (tid == 0) {
            unsigned t = sh_rank, cum = 0u;
            for (int d = 255; d >= 0; --d) {
                unsigned c = hist[d];
                if (cum + c > t) {
                    sh_prefix = pref | ((unsigned)d << shift);
                    sh_rank = t - cum;
                    break;
                }
                cum += c;
            }
        }
        __syncthreads();
    }

    // ---- Invert key map -> threshold float; stream masked row out ----
    const unsigned keyk = sh_prefix;
    const unsigned ub = (keyk & 0x80000000u) ? (keyk ^ 0x80000000u) : ~keyk;
    const float thr = __uint_as_float(ub);

    float* __restrict__ orow = out + rbase;
    #pragma unroll
    for (int i = 0; i < ITERS; ++i) {
        const int v = tid + i * BLOCK;
        float4 x = *reinterpret_cast<const float4*>(&row[v * 4]);
        float4 y;
        y.x = (x.x > thr) ? x.x : 0.0f;
        y.y = (x.y > thr) ? x.y : 0.0f;
        y.z = (x.z > thr) ? x.z : 0.0f;
        y.w = (x.w > thr) ? x.w : 0.0f;
        *reinterpret_cast<float4*>(&orow[v * 4]) = y;
    }
}

extern "C" void kernel_launch(void* const* d_in, const int* in_sizes, int n_in,
                              void* d_out, int out_size, void* d_ws, size_t ws_size,
                              hipStream_t stream) {
    const float* in = (const float*)d_in[0];
    const int* kp = (const int*)d_in[1];
    float* out = (float*)d_out;
    const int batch = in_sizes[0] / N_FEAT;            // 4096 rows
    KSparse_52879637348503_kernel<<<batch, BLOCK, 0, stream>>>(in, kp, out);
}